// Surv_Loss_73143293051388
// MI455X (gfx1250) — compile-verified
//
#include <hip/hip_runtime.h>

#define NROWS 8192
#define TCOLS 256
#define SIGMA 100.0f
#define ALPHA 1.0f

typedef __attribute__((ext_vector_type(2))) float v2f;
typedef __attribute__((ext_vector_type(8))) float v8f;

__global__ void init_kernel(float* __restrict__ S, float* __restrict__ out) {
    int i = threadIdx.x;
    if (i < TCOLS) S[i] = 0.0f;
    if (i == 0) out[0] = 0.0f;
}

// Phase 1: per 16-row tile, compute F1 = rowwise cumsum(y_pred) via triangular
// WMMA (f32 16x16x4), accumulate S[t] = sum_j ev[j]*[y[j]>t]*exp(F1[j,t]/sigma),
// and extract A[i] = F1[i, y[i]] through an LDS staging tile.
__launch_bounds__(256)
__global__ void phase1_kernel(const float* __restrict__ y_pred,
                              const int*   __restrict__ y,
                              const float* __restrict__ status,
                              float* __restrict__ S_glob,
                              float* __restrict__ A_glob) {
    __shared__ float S_lds[TCOLS];
    __shared__ float tile[8][16][16];

    const int tid  = threadIdx.x;
    const int wave = tid >> 5;
    const int lane = tid & 31;
    const int hi   = lane >> 4;          // half-wave select (K/M split)
    const int c    = lane & 15;          // column within tile (C/D layout)
    const int row_base = blockIdx.x * 128 + wave * 16;

    for (int i = tid; i < TCOLS; i += 256) S_lds[i] = 0.0f;
    __syncthreads();

    // Constant triangular B fragments: B[k][c] = (k <= c) within a 16x16 block.
    // B 4x16 layout: lanes 0-15 hold K rows {4s,4s+1}, lanes 16-31 hold {4s+2,4s+3}.
    v2f Btri[4];
#pragma unroll
    for (int s = 0; s < 4; ++s) {
        int k0 = 4 * s + (hi ? 2 : 0);
        Btri[s].x = (k0     <= c) ? 1.0f : 0.0f;
        Btri[s].y = (k0 + 1 <= c) ? 1.0f : 0.0f;
    }

    // Row metadata for the 8 M-rows this lane's C-fragment covers (M = v + 8*hi).
    int   y_rows[8];
    float ev01[8];                       // 1.0 if event row, else 0.0
#pragma unroll
    for (int v = 0; v < 8; ++v) {
        int gr = row_base + v + 8 * hi;
        y_rows[v] = y[gr];
        ev01[v]   = (status[gr] > 0.5f) ? 1.0f : 0.0f;
    }

    const int arow = row_base + c;                   // row for A-extraction (lanes 0-15)
    const int aty  = (hi == 0) ? y[arow] : 0;

    v8f carry = {};                                   // running cumsum at block entry

    for (int n = 0; n < 16; ++n) {
        const int colbase = n * 16;
        v8f C = carry;

        // In-block prefix sum: 4 accumulating WMMAs against triangular B.
#pragma unroll
        for (int s = 0; s < 4; ++s) {
            // A 16x4 layout: lane M = c; lanes 0-15 load K={0,1}, 16-31 load K={2,3}.
            const int kcol = colbase + 4 * s + (hi ? 2 : 0);
            const float* ap = y_pred + (size_t)(row_base + c) * TCOLS + kcol;
            v2f Af = *(const v2f*)ap;
            C = __builtin_amdgcn_wmma_f32_16x16x4_f32(
                    /*neg_a=*/false, Af, /*neg_b=*/false, Btri[s],
                    /*c_mod=*/(short)0, C, /*reuse_a=*/false, /*reuse_b=*/false);
        }

        // Carry update: broadcast column-15 values (lane 15 holds M=0..7,
        // lane 31 holds M=8..15) via v_readlane (pure VALU/SALU, no LDS).
        v8f newcarry;
#pragma unroll
        for (int v = 0; v < 8; ++v) {
            int lo = __builtin_amdgcn_readlane(__float_as_int(C[v]), 15);
            int hh = __builtin_amdgcn_readlane(__float_as_int(C[v]), 31);
            newcarry[v] = __int_as_float(hi ? hh : lo);
        }
        carry = newcarry;

        // S[t] partial: column t = colbase + c; rows M = v + 8*hi.
        // Branchless: unconditional v_exp, mask folded via cndmask + fma.
        {
            const int t = colbase + c;
            float part = 0.0f;
#pragma unroll
            for (int v = 0; v < 8; ++v) {
                float e   = __expf(C[v] * (1.0f / SIGMA));
                float msk = (y_rows[v] > t) ? ev01[v] : 0.0f;
                part = fmaf(msk, e, part);
            }
            atomicAdd(&S_lds[t], part);
        }

        // Stage F1 tile to LDS so lanes 0-15 can gather A[i] = F1[i, y[i]].
#pragma unroll
        for (int v = 0; v < 8; ++v)
            tile[wave][v + 8 * hi][c] = C[v];
        asm volatile("s_wait_dscnt 0" ::: "memory");   // same-wave LDS RAW order
        if (hi == 0 && (aty >> 4) == n)
            A_glob[arow] = tile[wave][c][aty & 15];
        asm volatile("" ::: "memory");
    }

    __syncthreads();
    for (int i = tid; i < TCOLS; i += 256)
        atomicAdd(&S_glob[i], S_lds[i]);
}

// Phase 2: per-row L1 term + alpha * ev * exp(-A/sigma) * S[y], reduce to scalar.
__launch_bounds__(256)
__global__ void phase2_kernel(const float* __restrict__ y_pred,
                              const int*   __restrict__ y,
                              const float* __restrict__ status,
                              const float* __restrict__ S_glob,
                              const float* __restrict__ A_glob,
                              float* __restrict__ out) {
    __shared__ float red[256];
    const int i = blockIdx.x * 256 + threadIdx.x;
    float val = 0.0f;
    if (i < NROWS) {
        const int   t  = y[i];
        const float st = status[i];
        const float p  = y_pred[(size_t)i * TCOLS + t];
        const float A  = A_glob[i];
        float l1 = -st * __logf(p) - (1.0f - st) * __logf(1.0f - A);
        float ev = (st > 0.5f) ? 1.0f : 0.0f;
        float l2 = ALPHA * ev * __expf(-A * (1.0f / SIGMA)) * S_glob[t];
        val = l1 + l2;
    }
    red[threadIdx.x] = val;
    __syncthreads();
    for (int s = 128; s > 0; s >>= 1) {
        if (threadIdx.x < s) red[threadIdx.x] += red[threadIdx.x + s];
        __syncthreads();
    }
    if (threadIdx.x == 0) atomicAdd(out, red[0]);
}

extern "C" void kernel_launch(void* const* d_in, const int* in_sizes, int n_in,
                              void* d_out, int out_size, void* d_ws, size_t ws_size,
                              hipStream_t stream) {
    const float* y_pred = (const float*)d_in[0];
    const int*   yv     = (const int*)d_in[1];
    const float* status = (const float*)d_in[2];
    float* out = (float*)d_out;
    float* S   = (float*)d_ws;           // 256 floats
    float* A   = S + TCOLS;              // 8192 floats

    init_kernel<<<1, 256, 0, stream>>>(S, out);
    phase1_kernel<<<NROWS / 128, 256, 0, stream>>>(y_pred, yv, status, S, A);
    phase2_kernel<<<NROWS / 256, 256, 0, stream>>>(y_pred, yv, status, S, A, out);
}